// BCosGCN_36429912604732
// MI455X (gfx1250) — compile-verified
//
#include <hip/hip_runtime.h>
#include <hip/hip_bf16.h>
#include <stdint.h>

typedef float v2f __attribute__((ext_vector_type(2)));
typedef float v8f __attribute__((ext_vector_type(8)));

#define N_NODES  100000
#define N_EDGES  800000
#define CH       128          // IN_CH == HIDDEN == 128 (both scatter inputs are 128-wide)
#define HIDDEN   128
#define OUT_CH   64
#define EPS_F    1e-6f
#define NORM_EPS 1e-12f

// ---------------------------------------------------------------------------
// Edge scatter: dst[row[e]] += src[col[e]] * w[e]      (128 channels / edge)
// One wave32 per edge, each lane owns 4 contiguous channels (float4).
// unsafeAtomicAdd -> global_atomic_add_f32 (L2-resident: 51.2MB << 192MB L2).
// ---------------------------------------------------------------------------
__global__ void edge_scatter_kernel(const float* __restrict__ src,
                                    const long long* __restrict__ row,
                                    const long long* __restrict__ col,
                                    const float* __restrict__ ew,
                                    float* __restrict__ dst)
{
    int gid  = blockIdx.x * blockDim.x + threadIdx.x;
    int e    = gid >> 5;
    int lane = gid & 31;
    if (e >= N_EDGES) return;

    long long r = row[e];
    long long c = col[e];
    float     w = ew[e];

    const float4 v = *(const float4*)(src + (size_t)c * CH + lane * 4);
    float* p = dst + (size_t)r * CH + lane * 4;
    unsafeAtomicAdd(p + 0, v.x * w);
    unsafeAtomicAdd(p + 1, v.y * w);
    unsafeAtomicAdd(p + 2, v.z * w);
    unsafeAtomicAdd(p + 3, v.w * w);
}

// ---------------------------------------------------------------------------
// Row L2-norms over 128 columns. One wave per row, shuffle-tree reduction.
// ---------------------------------------------------------------------------
__global__ void rownorm128_kernel(const float* __restrict__ A,
                                  float* __restrict__ out, int rows)
{
    int gid  = blockIdx.x * blockDim.x + threadIdx.x;
    int r    = gid >> 5;
    int lane = gid & 31;
    if (r >= rows) return;

    float4 v = *(const float4*)(A + (size_t)r * 128 + lane * 4);
    float s = v.x * v.x + v.y * v.y + v.z * v.z + v.w * v.w;
#pragma unroll
    for (int m = 16; m > 0; m >>= 1) s += __shfl_xor(s, m, 32);
    if (lane == 0) out[r] = sqrtf(s);
}

// ---------------------------------------------------------------------------
// BCos linear via V_WMMA_F32_16X16X4_F32 (fp32 in, fp32 accum — exact match
// to the fp32 reference; total GEMM work is only ~5 GFLOP so precision wins).
//
// Wave w of the block computes the 16x16 tile (rows blockIdx.x*16, cols w*16).
// A(16x4 f32) layout: lane L -> row L&15, K-pair (L>>4)*2  => contiguous
// float2 from row-major Z. B(4x16) is the mirror: lane L -> col L&15, same
// K-pair => contiguous float2 from row-major W ( = W^T tile ).
// C/D: lane L -> col n0+(L&15), VGPR v -> row m0 + v + 8*(L>>4).
//
// Epilogue: lin * max( lin/(max(rn,eps)*max(wn,eps)), EPS ) [+ ReLU].
// ---------------------------------------------------------------------------
__global__ void bcos_wmma_kernel(const float* __restrict__ Z,
                                 const float* __restrict__ W,
                                 const float* __restrict__ rn,
                                 const float* __restrict__ wn,
                                 float* __restrict__ Out,
                                 int n_out, int do_relu)
{
    const int lane  = threadIdx.x & 31;
    const int wave  = threadIdx.x >> 5;
    const int m0    = blockIdx.x * 16;
    const int n0    = wave * 16;
    const int idx15 = lane & 15;
    const int khalf = lane >> 4;          // 0: K={0,1}, 1: K={2,3} per step

    const float* arow = Z + (size_t)(m0 + idx15) * 128 + khalf * 2;
    const float* brow = W + (size_t)(n0 + idx15) * 128 + khalf * 2;

    v8f acc = {};
#pragma unroll
    for (int k = 0; k < 128; k += 4) {
        v2f a = *(const v2f*)(arow + k);
        v2f b = *(const v2f*)(brow + k);
        // (neg_a, A, neg_b, B, c_mod, C, reuse_a, reuse_b)
        acc = __builtin_amdgcn_wmma_f32_16x16x4_f32(
                  false, a, false, b, (short)0, acc, false, false);
    }

    const int   n   = n0 + idx15;
    const float wnv = fmaxf(wn[n], NORM_EPS);
#pragma unroll
    for (int v = 0; v < 8; ++v) {
        int   m    = m0 + v + 8 * khalf;
        float lin  = acc[v];
        float rnv  = fmaxf(rn[m], NORM_EPS);
        float cosv = fmaxf(lin / (rnv * wnv), EPS_F);   // clip(cos, EPS); ^(B-1)=^1
        float val  = lin * cosv;
        if (do_relu) val = fmaxf(val, 0.0f);
        Out[(size_t)m * n_out + n] = val;
    }
}

// ---------------------------------------------------------------------------
// Launch: z = scatter(x); h = relu(bcos(z,W1)); z2 = scatter(h);
//         out = bcos(z2,W2).   z-buffer is reused for z2.
// ---------------------------------------------------------------------------
extern "C" void kernel_launch(void* const* d_in, const int* in_sizes, int n_in,
                              void* d_out, int out_size, void* d_ws, size_t ws_size,
                              hipStream_t stream)
{
    const float*     x  = (const float*)d_in[0];
    const long long* ei = (const long long*)d_in[1];   // [2, E] int64
    const float*     ew = (const float*)d_in[2];
    const float*     W1 = (const float*)d_in[3];       // [128,128]
    const float*     W2 = (const float*)d_in[4];       // [64,128]
    float*           out = (float*)d_out;              // [100000,64]

    const long long* row = ei;
    const long long* col = ei + N_EDGES;

    float* z   = (float*)d_ws;                         // [N,128] (also z2)
    float* h   = z   + (size_t)N_NODES * CH;           // [N,128]
    float* rnb = h   + (size_t)N_NODES * CH;           // [N]
    float* wn1 = rnb + N_NODES;                        // [128]
    float* wn2 = wn1 + HIDDEN;                         // [64]

    const size_t zbytes = (size_t)N_NODES * CH * sizeof(float);

    const int scatter_blocks = (N_EDGES * 32) / 256;       // 100000
    const int node_norm_blocks = (N_NODES * 32) / 256;     // 12500
    const int gemm_blocks = N_NODES / 16;                  // 6250

    // ---- layer 1 ----
    hipMemsetAsync(z, 0, zbytes, stream);
    edge_scatter_kernel<<<scatter_blocks, 256, 0, stream>>>(x, row, col, ew, z);
    rownorm128_kernel<<<node_norm_blocks, 256, 0, stream>>>(z, rnb, N_NODES);
    rownorm128_kernel<<<(HIDDEN * 32) / 256, 256, 0, stream>>>(W1, wn1, HIDDEN);
    bcos_wmma_kernel<<<gemm_blocks, 32 * (HIDDEN / 16), 0, stream>>>(
        z, W1, rnb, wn1, h, HIDDEN, 1);

    // ---- layer 2 ----
    hipMemsetAsync(z, 0, zbytes, stream);
    edge_scatter_kernel<<<scatter_blocks, 256, 0, stream>>>(h, row, col, ew, z);
    rownorm128_kernel<<<node_norm_blocks, 256, 0, stream>>>(z, rnb, N_NODES);
    rownorm128_kernel<<<(OUT_CH * 32) / 256, 256, 0, stream>>>(W2, wn2, OUT_CH);
    bcos_wmma_kernel<<<gemm_blocks, 32 * (OUT_CH / 16), 0, stream>>>(
        z, W2, rnb, wn2, out, OUT_CH, 0);
}